// minGRU2_12575664243302
// MI455X (gfx1250) — compile-verified
//
#include <hip/hip_runtime.h>
#include <stdint.h>

#define THREADS 256
#define L_SEQ   8192
#define TILE    4096
#define NT      (L_SEQ / TILE)       // 2 tiles, scalar carry between them
#define EPT     (TILE / THREADS)     // 16 elements per thread per tile
#define CPT     (TILE / 4 / THREADS) // 4 float4 chunks per thread per array
#define PADW(w) ((w) + (((w) >> 5) << 2))   // +4 words per 32 words: 16B-aligned chunks, few bank conflicts
#define LDSW    (TILE + (TILE >> 5) * 4)    // 4608 words per staged array

#if __has_builtin(__builtin_amdgcn_global_load_async_to_lds_b128) && \
    __has_builtin(__builtin_amdgcn_global_store_async_from_lds_b128)
#define HAVE_ASYNC 1
#else
#define HAVE_ASYNC 0
#endif

typedef int v4i __attribute__((vector_size(16)));
typedef __attribute__((address_space(1))) v4i* gbl_v4i_p;   // prints as "v4i __device__ *"
typedef __attribute__((address_space(3))) v4i* lds_v4i_p;   // prints as "v4i __shared__ *"

// Generic->LDS: low 32 bits of a generic shared pointer are the LDS offset (aperture form).
__device__ __forceinline__ lds_v4i_p to_lds(const void* p) {
  return (lds_v4i_p)(uint32_t)(uintptr_t)p;
}
// Generic->global: same 64-bit VA.
__device__ __forceinline__ gbl_v4i_p to_gbl(const void* p) {
  return (gbl_v4i_p)(uintptr_t)p;
}

__device__ __forceinline__ void wait_async0() {
#if __has_builtin(__builtin_amdgcn_s_wait_asynccnt)
  __builtin_amdgcn_s_wait_asynccnt(0);
#else
  asm volatile("s_wait_asynccnt 0" ::: "memory");
#endif
}

__global__ __launch_bounds__(THREADS) void mingru_scan_kernel(
    const float* __restrict__ x, float* __restrict__ out) {
  __shared__ __align__(16) float hbuf[LDSW];   // h tile -> per-element a -> result
  __shared__ __align__(16) float gbuf[LDSW];   // g tile -> per-element b
  __shared__ float sA[THREADS];
  __shared__ float sB[THREADS];
  __shared__ float sCarry;

  const int tid = threadIdx.x;
  const int s   = blockIdx.x;        // 0 .. B*256-1
  const int b   = s >> 8;
  const int c   = s & 255;           // output channel
  const bool fwd = (c < 128);
  const int hc = fwd ? c         : (128 + c);  // bwd: x channel 256 + (c-128)
  const int gc = fwd ? (128 + c) : (256 + c);  // bwd: x channel 384 + (c-128)

  const float* __restrict__ hsrc = x   + (size_t)(b * 512 + hc) * L_SEQ;
  const float* __restrict__ gsrc = x   + (size_t)(b * 512 + gc) * L_SEQ;
  float*       __restrict__ odst = out + (size_t)(b * 256 + c)  * L_SEQ;

  if (tid == 0) sCarry = 0.f;   // h_{-1} = 0

  for (int ti = 0; ti < NT; ++ti) {
    // memory-order tile base; scan order walks it forward (fwd) or backward (bwd)
    const int tb = fwd ? (ti * TILE) : (L_SEQ - (ti + 1) * TILE);

#if HAVE_ASYNC
    if (ti > 0) wait_async0();   // prior tile's async stores still read hbuf
#endif
    if (ti > 0) __syncthreads();

    // ---- Phase A: stage h,g tile into LDS (coalesced B128, async on CDNA5) ----
#pragma unroll
    for (int i = 0; i < CPT; ++i) {
      const int k    = i * THREADS + tid;       // chunk id: lanes contiguous -> 512B/wave
      const int lofs = 16 * k + 16 * (k >> 3);  // padded, 16B-aligned LDS byte offset
#if HAVE_ASYNC
      __builtin_amdgcn_global_load_async_to_lds_b128(
          to_gbl(hsrc + tb + 4 * k), to_lds((const char*)hbuf + lofs), 0, 0);
      __builtin_amdgcn_global_load_async_to_lds_b128(
          to_gbl(gsrc + tb + 4 * k), to_lds((const char*)gbuf + lofs), 0, 0);
#else
      *(float4*)((char*)hbuf + lofs) = *(const float4*)(hsrc + tb + 4 * k);
      *(float4*)((char*)gbuf + lofs) = *(const float4*)(gsrc + tb + 4 * k);
#endif
    }
#if HAVE_ASYNC
    wait_async0();
#endif
    __syncthreads();
    const float hIn = sCarry;   // running h entering this tile

    // ---- Phase B: per-thread affine fold over 16 contiguous scan elements ----
    float A = 1.f, Bc = 0.f;
    const int e0 = tid * EPT;
#pragma unroll
    for (int j = 0; j < EPT; ++j) {
      const int el = e0 + j;
      const int ml = fwd ? el : (TILE - 1 - el);   // memory-local index
      const int w  = PADW(ml);
      const float h = hbuf[w];
      const float g = gbuf[w];
      const float t  = __expf(-fabsf(g));
      const float r  = 1.f / (1.f + t);
      const float a  = (g >= 0.f) ? (t * r) : r;   // a = sigmoid(-g) = 1 - z
      const float z  = 1.f - a;                    // z = sigmoid(g)
      const float ht = (h < 0.f) ? __expf(h) : (1.f + h);  // exp(log_h_tilde)
      const float bb = z * ht;
      hbuf[w] = a;              // cache per-element coefficients for replay
      gbuf[w] = bb;
      Bc = fmaf(a, Bc, bb);     // compose: later ∘ earlier
      A *= a;
    }
    sA[tid] = A;
    sB[tid] = Bc;
    __syncthreads();

    // ---- Phase C: Hillis-Steele inclusive scan of 256 affine pairs ----
#pragma unroll
    for (int off = 1; off < THREADS; off <<= 1) {
      float pA = 1.f, pB = 0.f;
      const float mA = sA[tid], mB = sB[tid];
      if (tid >= off) { pA = sA[tid - off]; pB = sB[tid - off]; }
      __syncthreads();
      if (tid >= off) {
        sA[tid] = mA * pA;            // (mA,mB) ∘ (pA,pB)
        sB[tid] = fmaf(mA, pB, mB);
      }
      __syncthreads();
    }

    // ---- Phase D: replay with exclusive prefix; write results in memory order ----
    float hrun = (tid == 0) ? hIn : fmaf(sA[tid - 1], hIn, sB[tid - 1]);
#pragma unroll
    for (int j = 0; j < EPT; ++j) {
      const int el = e0 + j;
      const int ml = fwd ? el : (TILE - 1 - el);
      const int w  = PADW(ml);
      hrun = fmaf(hbuf[w], hrun, gbuf[w]);
      hbuf[w] = hrun;           // result staged for coalesced store
    }
    if (tid == THREADS - 1) sCarry = hrun;   // carry into next tile
    __syncthreads();

    // ---- Phase E: coalesced B128 store of the tile (async LDS->global) ----
#pragma unroll
    for (int i = 0; i < CPT; ++i) {
      const int k    = i * THREADS + tid;
      const int lofs = 16 * k + 16 * (k >> 3);
#if HAVE_ASYNC
      __builtin_amdgcn_global_store_async_from_lds_b128(
          to_gbl(odst + tb + 4 * k), to_lds((const char*)hbuf + lofs), 0, 0);
#else
      *(float4*)(odst + tb + 4 * k) = *(const float4*)((const char*)hbuf + lofs);
#endif
    }
  }
  // s_endpgm performs an implicit wait-idle, draining pending async stores.
}

extern "C" void kernel_launch(void* const* d_in, const int* in_sizes, int n_in,
                              void* d_out, int out_size, void* d_ws, size_t ws_size,
                              hipStream_t stream) {
  (void)n_in; (void)out_size; (void)d_ws; (void)ws_size;
  const float* x = (const float*)d_in[0];
  float* out = (float*)d_out;
  // sequences = B * 256 (128 fwd + 128 bwd output channels per batch)
  const int nseq = in_sizes[0] / (2 * L_SEQ);
  mingru_scan_kernel<<<dim3(nseq), dim3(THREADS), 0, stream>>>(x, out);
}